// Prior_SPDE_83829171683377
// MI455X (gfx1250) — compile-verified
//
#include <hip/hip_runtime.h>

#define NN 1024      // nodes per time slab (32x32)
#define NT 8         // time steps

typedef __attribute__((ext_vector_type(2))) float v2f;
typedef __attribute__((ext_vector_type(4))) float v4f;
typedef __attribute__((ext_vector_type(8))) float v8f;

// Stencil offsets in linear node index, slot order:
//   0:-33  1:-32  2:-31  3:-1  4:0  5:+1  6:+31  7:+32  8:+33
struct Coefs { float c[9]; float w; };

__device__ __forceinline__ Coefs stencil_coefs(const float* __restrict__ kap,
                                               const float* __restrict__ mm,
                                               const float* __restrict__ HH,
                                               const float* __restrict__ tt,
                                               int b, int t, int node) {
  // input layouts (innermost = t, stride NT): kappa[b,1,node,t], m[b,2,node,t],
  // H[b,2,2,node,t], tau[b,1,node,t]
  float kp  = kap[(b * NN + node) * NT + t];
  float m1  = mm[((2 * b + 0) * NN + node) * NT + t];
  float m2  = mm[((2 * b + 1) * NN + node) * NT + t];
  float h11 = HH[((4 * b + 0) * NN + node) * NT + t];
  float h12 = HH[((4 * b + 1) * NN + node) * NT + t];
  float h22 = HH[((4 * b + 3) * NN + node) * NT + t];
  float ta  = tt[(b * NN + node) * NT + t];
  int ii = node >> 5, jj = node & 31;
  bool up = ii > 0, dn = ii < 31, lf = jj > 0, rt = jj < 31;
  float cxy = 0.5f * h12;
  Coefs R;
  R.c[4] = kp * kp + 2.0f * h11 + 2.0f * h22 + 1.0f;     // M = I + dt*A diag, dt=1
  R.c[5] = rt ? (-h11 + 0.5f * m1) : 0.0f;               // (0,+1)
  R.c[3] = lf ? (-h11 - 0.5f * m1) : 0.0f;               // (0,-1)
  R.c[7] = dn ? (-h22 + 0.5f * m2) : 0.0f;               // (+1,0)
  R.c[1] = up ? (-h22 - 0.5f * m2) : 0.0f;               // (-1,0)
  R.c[8] = (dn && rt) ? -cxy : 0.0f;                     // (+1,+1)
  R.c[0] = (up && lf) ? -cxy : 0.0f;                     // (-1,-1)
  R.c[6] = (dn && lf) ?  cxy : 0.0f;                     // (+1,-1)
  R.c[2] = (up && rt) ?  cxy : 0.0f;                     // (-1,+1)
  R.w = 1.0f / (ta * ta);                                // dt / tau^2
  return R;
}

// ---------------- kernel 1: wide NT zero fill of the whole output ----------------
__global__ __launch_bounds__(256) void spde_zero(v4f* __restrict__ p, long n4) {
  long i = (long)blockIdx.x * blockDim.x + threadIdx.x;
  long stride = (long)gridDim.x * blockDim.x;
  v4f z = {0.0f, 0.0f, 0.0f, 0.0f};
  for (; i < n4; i += stride) __builtin_nontemporal_store(z, p + i);
}

// ---------------- kernel 2: band tiles of D = M^T diag(w) M (+ diag(w_next)) -----
// One block per (b, t, 16-row block J). Dense LDS slab of M:
//   rows    i  in [J-33, J+50]  (84 rows, 21 WMMA K-steps of 4)
//   columns k  in [J-80, J+96)  (176 cols, stride 180)
// A(16x4)[M=lm, K=v+2*half] = w[ip] * Ms[ip][80+lm]     (= (WM)^T slab)
// B(4x16)[K=v+2*half, N=lm] =         Ms[ip][16*ct+lm]
// C/D(16x16)[M=v+8*half, N=lm]
__global__ __launch_bounds__(256) void spde_band_wmma(const float* __restrict__ kap,
                                                      const float* __restrict__ mm,
                                                      const float* __restrict__ HH,
                                                      const float* __restrict__ tt,
                                                      float* __restrict__ out) {
  __shared__ __align__(16) float Ms[84 * 180];   // 60480 B
  __shared__ float sW[84];
  int blk = blockIdx.x;
  int jb = blk & 63;
  int t  = (blk >> 6) & 7;
  int b  = blk >> 9;
  int J  = jb << 4;
  int tid = threadIdx.x;

  // cooperative zero of the slab
  v4f* Mz = (v4f*)Ms;
  v4f z4 = {0.0f, 0.0f, 0.0f, 0.0f};
  for (int i = tid; i < (84 * 180) / 4; i += 256) Mz[i] = z4;
  if (tid < 84) sW[tid] = 0.0f;
  __syncthreads();

  // each of the first 84 threads owns one row of the slab
  if (tid < 84) {
    int i = J - 33 + tid;
    if (i >= 0 && i < NN) {
      Coefs R = stencil_coefs(kap, mm, HH, tt, b, t, i);
      const int delta[9] = { -33, -32, -31, -1, 0, 1, 31, 32, 33 };
#pragma unroll
      for (int c = 0; c < 9; ++c)
        Ms[tid * 180 + tid + delta[c] + 47] = R.c[c];   // col = (i+d)-(J-80) in [14,163]
      sW[tid] = R.w;
    }
  }
  __syncthreads();

  int lane = tid & 31;
  int wv   = tid >> 5;                 // 8 waves per block
  int half = lane >> 4;                // 0: lanes 0-15, 1: lanes 16-31
  int lm   = lane & 15;
  float* Dbase = out + (((size_t)(b * 3 + 0) * NT + t) << 20);   // N*N = 1<<20

  for (int ct = wv; ct < 11; ct += 8) {          // wave-uniform tile loop
    int K = J + ct * 16 - 80;
    if (K < 0 || K >= NN) continue;              // uniform per wave
    v8f acc = {0.0f, 0.0f, 0.0f, 0.0f, 0.0f, 0.0f, 0.0f, 0.0f};
#pragma unroll
    for (int s = 0; s < 21; ++s) {
      int ip0 = 4 * s + 2 * half;                // local row for fragment element v=0
      int ip1 = ip0 + 1;                         // v=1
      v2f a, bb;
      a[0]  = sW[ip0] * Ms[ip0 * 180 + 80 + lm];
      a[1]  = sW[ip1] * Ms[ip1 * 180 + 80 + lm];
      bb[0] = Ms[ip0 * 180 + ct * 16 + lm];
      bb[1] = Ms[ip1 * 180 + ct * 16 + lm];
      acc = __builtin_amdgcn_wmma_f32_16x16x4_f32(false, a, false, bb,
                                                  (short)0, acc, false, false);
    }
    if (K == J && t < NT - 1) {                  // diagonal tile: += w_{t+1}[j]
      float tn = tt[(b * NN + (J + lm)) * NT + (t + 1)];
      float wn = 1.0f / (tn * tn);
#pragma unroll
      for (int v = 0; v < 8; ++v)
        if (lm == v + 8 * half) acc[v] += wn;
    }
#pragma unroll
    for (int v = 0; v < 8; ++v) {                // C/D: (M=v+8*half, N=lm)
      int j = J + v + 8 * half;
      __builtin_nontemporal_store(acc[v], &Dbase[(size_t)j * NN + K + lm]);
    }
  }
}

// ---------------- kernel 3: scatter lower/upper (9 nonzeros per row) -------------
// lower[b,t][r+d, r]   = -w_t[r] * M_t[r, r+d]      (t = 1..7)
// upper[b,t-1][r, r+d] = -w_t[r] * M_t[r, r+d]      (t = 1..7)
__global__ __launch_bounds__(256) void spde_scatter_lu(const float* __restrict__ kap,
                                                       const float* __restrict__ mm,
                                                       const float* __restrict__ HH,
                                                       const float* __restrict__ tt,
                                                       float* __restrict__ out, int total) {
  int id = blockIdx.x * blockDim.x + threadIdx.x;
  if (id >= total) return;
  int r = id & (NN - 1);
  int t = (id >> 10) & (NT - 1);
  int b = id >> 13;
  if (t == 0) return;
  Coefs R = stencil_coefs(kap, mm, HH, tt, b, t, r);
  int ii = r >> 5, jj = r & 31;
  bool up = ii > 0, dn = ii < 31, lf = jj > 0, rt = jj < 31;
  const bool valid[9] = { up && lf, up, up && rt, lf, true, rt, dn && lf, dn, dn && rt };
  const int  delta[9] = { -33, -32, -31, -1, 0, 1, 31, 32, 33 };
  float* lowerB = out + (((size_t)(b * 3 + 1) * NT + t) << 20);
  float* upperB = out + (((size_t)(b * 3 + 2) * NT + (t - 1)) << 20);
#pragma unroll
  for (int c = 0; c < 9; ++c) {
    if (valid[c]) {
      float v = -R.w * R.c[c];
      int q = r + delta[c];
      __builtin_nontemporal_store(v, &lowerB[(size_t)q * NN + r]);
      __builtin_nontemporal_store(v, &upperB[(size_t)r * NN + q]);
    }
  }
}

extern "C" void kernel_launch(void* const* d_in, const int* in_sizes, int n_in,
                              void* d_out, int out_size, void* d_ws, size_t ws_size,
                              hipStream_t stream) {
  const float* kap = (const float*)d_in[0];
  const float* mm  = (const float*)d_in[1];
  const float* HH  = (const float*)d_in[2];
  const float* tt  = (const float*)d_in[3];
  float* out = (float*)d_out;
  int nb = in_sizes[0] / (NN * NT);              // kappa is [nb,1,N,NT]

  long n4 = (long)out_size / 4;                  // out_size = nb*3*8*N*N, /4 exact
  spde_zero<<<12288, 256, 0, stream>>>((v4f*)d_out, n4);

  spde_band_wmma<<<nb * NT * 64, 256, 0, stream>>>(kap, mm, HH, tt, out);

  int total = nb * NT * NN;
  spde_scatter_lu<<<(total + 255) / 256, 256, 0, stream>>>(kap, mm, HH, tt, out, total);
}